// RetinaNetPostProcess_83021717832156
// MI455X (gfx1250) — compile-verified
//
#include <hip/hip_runtime.h>
#include <math.h>

#define A_ANCH 9
#define C_CLS 80
#define AC 720            // A*C
#define TOPK 1000
#define B_IMG 8
#define NLVL 3
#define NCAND 3000        // 3 levels * 1000
#define NPAD 3072         // padded to multiple of 16
#define NSORT 4096        // bitonic power-of-two
#define NWORDS 96         // 3072 / 32
#define NTILES 192        // 3072 / 16
#define DETS 100
#define NMS_TH 0.5f
#define SCORE_TH 0.05f
#define BBOX_CLIP 4.135166556742356f
#define CLS_OFF 4096.0f
#define NBINS 2048
#define TIE_CAP 4096

typedef __attribute__((ext_vector_type(2))) float v2f;
typedef __attribute__((ext_vector_type(8))) float v8f;

__device__ __forceinline__ int bin_of(float f) {
    int b = (int)floorf((f + 16.0f) * 64.0f);
    return b < 0 ? 0 : (b > (NBINS - 1) ? (NBINS - 1) : b);
}

// ---------------- Stage 1a: per-(image,level) histogram of logits ----------------
__global__ void hist_kernel(const float* __restrict__ logits, int perImg,
                            int* __restrict__ histBase /* + lvl*8*NBINS */) {
    __shared__ int h[NBINS];
    const int b = blockIdx.y;
    const int tid = threadIdx.x;
    for (int i = tid; i < NBINS; i += blockDim.x) h[i] = 0;
    __syncthreads();
    const float4* p = (const float4*)(logits + (size_t)b * perImg);
    const int n4 = perImg >> 2;
    const int stride = gridDim.x * blockDim.x;
    for (int i = blockIdx.x * blockDim.x + tid; i < n4; i += stride) {
        __builtin_prefetch(&p[i + stride], 0, 1);   // -> global_prefetch_b8
        float4 v = p[i];
        atomicAdd(&h[bin_of(v.x)], 1);
        atomicAdd(&h[bin_of(v.y)], 1);
        atomicAdd(&h[bin_of(v.z)], 1);
        atomicAdd(&h[bin_of(v.w)], 1);
    }
    __syncthreads();
    int* gh = histBase + (size_t)b * NBINS;
    for (int i = tid; i < NBINS; i += blockDim.x)
        if (h[i]) atomicAdd(&gh[i], h[i]);
}

// ---------------- Stage 1b: find threshold bin, strictly-greater count, tie quota ----
__global__ void thresh_kernel(const int* __restrict__ hist, int* __restrict__ meta) {
    const int bl = blockIdx.x;   // bl = lvl*8 + b, 24 total
    if (threadIdx.x != 0) return;
    const int* h = hist + (size_t)bl * NBINS;
    int total = 0, t = 0, G = 0;
    for (int i = NBINS - 1; i >= 0; --i) {
        int c = h[i];
        if (total + c >= TOPK) { t = i; G = total; break; }
        total += c;
        if (i == 0) { t = 0; G = total; }
    }
    int quota = TOPK - G;
    if (quota < 0) quota = 0;
    meta[bl * 4 + 0] = t;
    meta[bl * 4 + 1] = G;
    meta[bl * 4 + 2] = quota;
}

// ---------------- Stage 1c: compact strictly-above-threshold + gather tie pool -------
__global__ void compact_kernel(const float* __restrict__ logits, int perImg, int lvl,
                               const int* __restrict__ meta, int* __restrict__ cnt,
                               int* __restrict__ cand, int* __restrict__ tieBuf) {
    const int b = blockIdx.y;
    const int bl = lvl * 8 + b;
    const int t = meta[bl * 4 + 0];
    const float4* p = (const float4*)(logits + (size_t)b * perImg);
    const int n4 = perImg >> 2;
    const int stride = gridDim.x * blockDim.x;
    int* candB = cand + (size_t)bl * 1024;
    int* tieB  = tieBuf + (size_t)bl * TIE_CAP;
    for (int i = blockIdx.x * blockDim.x + threadIdx.x; i < n4; i += stride) {
        float4 v = p[i];
        float vv[4] = {v.x, v.y, v.z, v.w};
#pragma unroll
        for (int c = 0; c < 4; ++c) {
            int bn = bin_of(vv[c]);
            if (bn > t) {
                int pos = atomicAdd(&cnt[bl * 2 + 0], 1);
                if (pos < TOPK) candB[pos] = i * 4 + c;
            } else if (bn == t) {
                int tp = atomicAdd(&cnt[bl * 2 + 1], 1);
                if (tp < TIE_CAP) tieB[tp] = i * 4 + c;
            }
        }
    }
}

// ------- Stage 1d: deterministic tie-break — quota smallest flat indices ------------
__global__ __launch_bounds__(1024)
void tie_kernel(const int* __restrict__ meta, const int* __restrict__ cnt,
                const int* __restrict__ tieBuf, int* __restrict__ cand) {
    __shared__ int key[TIE_CAP];
    const int bl = blockIdx.x, tid = threadIdx.x;
    const int G = meta[bl * 4 + 1];
    const int quota = meta[bl * 4 + 2];
    int T = cnt[bl * 2 + 1];
    if (T > TIE_CAP) T = TIE_CAP;
    for (int i = tid; i < TIE_CAP; i += blockDim.x)
        key[i] = (i < T) ? tieBuf[(size_t)bl * TIE_CAP + i] : 0x7FFFFFFF;
    for (int k2 = 2; k2 <= TIE_CAP; k2 <<= 1) {
        for (int j2 = k2 >> 1; j2 > 0; j2 >>= 1) {
            __syncthreads();
            for (int tt = tid; tt < TIE_CAP / 2; tt += blockDim.x) {
                int i  = (tt & ~(j2 - 1)) * 2 + (tt & (j2 - 1));
                int ix = i + j2;
                bool up = ((i & k2) == 0);
                int ki = key[i], kx = key[ix];
                if (up ? (ki > kx) : (ki < kx)) { key[i] = kx; key[ix] = ki; }
            }
        }
    }
    __syncthreads();
    for (int i = tid; i < quota; i += blockDim.x) {
        int v = key[i];
        if (v == 0x7FFFFFFF) v = key[0];   // degenerate guard (never expected)
        cand[(size_t)bl * 1024 + G + i] = v;
    }
}

// ---------------- Stage 2: decode selected candidates ----------------
__global__ void decode_kernel(const float* an0, const float* an1, const float* an2,
                              const int* co0, const int* co1, const int* co2,
                              const float* lg0, const float* lg1, const float* lg2,
                              const float* rg0, const float* rg1, const float* rg2,
                              const float* __restrict__ img, const int* __restrict__ cand,
                              float* __restrict__ boxes, float* __restrict__ scores,
                              int* __restrict__ labels) {
    int idx = blockIdx.x * blockDim.x + threadIdx.x;
    if (idx >= 24 * 1024) return;
    int bl = idx >> 10, j = idx & 1023;
    if (j >= TOPK) return;
    int lvl = bl / 8, b = bl % 8;
    const float *an, *lg, *rg; const int* co; int K, H, W;
    if (lvl == 0)      { an = an0; co = co0; lg = lg0; rg = rg0; K = 4096; H = 128; W = 128; }
    else if (lvl == 1) { an = an1; co = co1; lg = lg1; rg = rg1; K = 2048; H = 64;  W = 64;  }
    else               { an = an2; co = co2; lg = lg2; rg = rg2; K = 1024; H = 32;  W = 32;  }

    int e = cand[(size_t)bl * 1024 + j];
    int k  = e / AC;
    int r  = e - k * AC;
    int a  = r / C_CLS;
    int cl = r - a * C_CLS;

    int y = co[((size_t)b * K + k) * 2 + 0];
    int x = co[((size_t)b * K + k) * 2 + 1];

    size_t abase = (((size_t)b * (4 * A_ANCH) + (size_t)a * 4) * H + y) * W + x;
    size_t hw = (size_t)H * W;
    float x1 = an[abase], y1 = an[abase + hw], x2 = an[abase + 2 * hw], y2 = an[abase + 3 * hw];

    size_t rbase = ((size_t)b * K + k) * (A_ANCH * 4) + (size_t)a * 4;
    float dx = rg[rbase], dy = rg[rbase + 1];
    float dw = fminf(rg[rbase + 2], BBOX_CLIP);
    float dh = fminf(rg[rbase + 3], BBOX_CLIP);

    float w = x2 - x1, h = y2 - y1;
    float cx = x1 + 0.5f * w, cy = y1 + 0.5f * h;
    float pcx = dx * w + cx, pcy = dy * h + cy;
    float pw = expf(dw) * w, ph = expf(dh) * h;
    float ih = img[b * 2 + 0], iw = img[b * 2 + 1];
    float nx1 = fminf(fmaxf(pcx - 0.5f * pw, 0.0f), iw);
    float ny1 = fminf(fmaxf(pcy - 0.5f * ph, 0.0f), ih);
    float nx2 = fminf(fmaxf(pcx + 0.5f * pw, 0.0f), iw);
    float ny2 = fminf(fmaxf(pcy + 0.5f * ph, 0.0f), ih);

    float logit = lg[(size_t)b * K * AC + e];
    float s = 1.0f / (1.0f + expf(-logit));

    int d = lvl * TOPK + j;             // [0, 3000)
    size_t ob = (size_t)b * NPAD + d;
    boxes[ob * 4 + 0] = nx1; boxes[ob * 4 + 1] = ny1;
    boxes[ob * 4 + 2] = nx2; boxes[ob * 4 + 3] = ny2;
    scores[ob] = s;
    labels[ob] = cl;
}

// ---------------- Stage 3: per-image bitonic sort by score (desc) ----------------
__global__ __launch_bounds__(1024)
void sort_kernel(const float* __restrict__ scores, const float* __restrict__ boxes,
                 const int* __restrict__ labels,
                 float* __restrict__ sb, float* __restrict__ sob, float* __restrict__ ss,
                 int* __restrict__ sl, float* __restrict__ ar) {
    __shared__ float key[NSORT];
    __shared__ int   sid[NSORT];
    const int b = blockIdx.x, tid = threadIdx.x;
    for (int i = tid; i < NSORT; i += blockDim.x) {
        float s = (i < NCAND) ? scores[(size_t)b * NPAD + i] : -3.0e30f;
        key[i] = -s;                       // ascending sort on -score
        sid[i] = i;
    }
    for (int k2 = 2; k2 <= NSORT; k2 <<= 1) {
        for (int j2 = k2 >> 1; j2 > 0; j2 >>= 1) {
            __syncthreads();
            for (int t = tid; t < NSORT / 2; t += blockDim.x) {
                int i  = (t & ~(j2 - 1)) * 2 + (t & (j2 - 1));
                int ix = i + j2;
                bool up = ((i & k2) == 0);
                float ki = key[i], kx = key[ix];
                if (up ? (ki > kx) : (ki < kx)) {
                    key[i] = kx; key[ix] = ki;
                    int tm = sid[i]; sid[i] = sid[ix]; sid[ix] = tm;
                }
            }
        }
    }
    __syncthreads();
    for (int r = tid; r < NPAD; r += blockDim.x) {
        int j = sid[r];
        bool ok = (j < NCAND) && (key[r] < 1.0e30f);
        float s = ok ? -key[r] : -3.0e30f;
        float x1 = 0, y1 = 0, x2 = 0, y2 = 0; int lbl = 0;
        if (ok) {
            size_t sbase = ((size_t)b * NPAD + j) * 4;
            x1 = boxes[sbase]; y1 = boxes[sbase + 1];
            x2 = boxes[sbase + 2]; y2 = boxes[sbase + 3];
            lbl = labels[(size_t)b * NPAD + j];
        }
        float off = (float)lbl * CLS_OFF;
        size_t dbase = ((size_t)b * NPAD + r) * 4;
        sb[dbase] = x1; sb[dbase + 1] = y1; sb[dbase + 2] = x2; sb[dbase + 3] = y2;
        sob[dbase] = x1 + off; sob[dbase + 1] = y1 + off;
        sob[dbase + 2] = x2 + off; sob[dbase + 3] = y2 + off;
        ss[(size_t)b * NPAD + r] = s;
        sl[(size_t)b * NPAD + r] = lbl;
        ar[(size_t)b * NPAD + r] = (x2 - x1) * (y2 - y1);
    }
}

// ---------------- Stage 4: IoU > thresh suppression bitmask (WMMA area-sums) -------
__global__ __launch_bounds__(1024)
void supp_kernel(const float* __restrict__ sob, const float* __restrict__ ar,
                 unsigned* __restrict__ supp) {
    const int I = blockIdx.x;        // row tile
    const int b = blockIdx.y;
    const int tid = threadIdx.x;
    __shared__ float rb[16 * 4];
    __shared__ float ra[16];
    if (tid < 64) rb[tid] = sob[((size_t)b * NPAD + I * 16) * 4 + tid];
    if (tid < 16) ra[tid] = ar[(size_t)b * NPAD + I * 16 + tid];
    __syncthreads();

    const int wave = tid >> 5;
    const int lane = tid & 31;
    const int half = lane >> 4;          // 0: lanes 0-15, 1: lanes 16-31
    const float4* sob4 = (const float4*)sob;

    // A (16x4, K=2 live): col0 = row areas, col1 = 1 — invariant across J, hoisted.
    v2f av;
    av.x = (half == 0) ? ra[lane & 15] : 0.0f;
    av.y = (half == 0) ? 1.0f : 0.0f;
    const float bsel = (half == 0) ? 1.0f : 0.0f;

    for (int J = wave; J < NTILES; J += 32) {
        if (J < I) continue;             // strictly below diagonal: no j>i bits
        int col = J * 16 + (lane & 15);
        float4 cb = sob4[(size_t)b * NPAD + col];
        float ca = ar[(size_t)b * NPAD + col];

        // B (4x16, K=2 live): row0 = 1, row1 = col areas
        v2f bv;
        bv.x = bsel;
        bv.y = bsel * ca;
        v8f cz = {};
        // D[m][n] = area_row[m] + area_col[n]  on the matrix pipe
        v8f d = __builtin_amdgcn_wmma_f32_16x16x4_f32(
            false, av, false, bv, (short)0, cz, false, false);

#pragma unroll
        for (int v = 0; v < 8; ++v) {
            int m = v + (half << 3);
            int grow = I * 16 + m;
            float rx1 = rb[m * 4 + 0], ry1 = rb[m * 4 + 1];
            float rx2 = rb[m * 4 + 2], ry2 = rb[m * 4 + 3];
            float w = fminf(rx2, cb.z) - fmaxf(rx1, cb.x);
            float h = fminf(ry2, cb.w) - fmaxf(ry1, cb.y);
            w = fmaxf(w, 0.0f); h = fmaxf(h, 0.0f);
            float inter = w * h;
            float iou = inter / (d[v] - inter + 1e-9f);
            bool pred = (iou > NMS_TH) && (col > grow) && (col < NCAND) && (grow < NCAND);
            unsigned msk = __builtin_amdgcn_ballot_w32(pred);
            if (lane == 0) {
                unsigned lo = msk & 0xFFFFu, hi = msk >> 16;
                int word = J >> 1, sh = (J & 1) * 16;
                if (lo) atomicOr(&supp[((size_t)b * NPAD + I * 16 + v) * NWORDS + word], lo << sh);
                if (hi) atomicOr(&supp[((size_t)b * NPAD + I * 16 + v + 8) * NWORDS + word], hi << sh);
            }
        }
    }
}

// ---------------- Stage 5: sequential greedy resolve + emit top-100 ----------------
__global__ void resolve_kernel(const float* __restrict__ ss, const float* __restrict__ sb,
                               const int* __restrict__ sl, const unsigned* __restrict__ supp,
                               float* __restrict__ out) {
    const int b = blockIdx.x, tid = threadIdx.x;
    __shared__ unsigned rem[NWORDS];
    __shared__ int sflag, sdone, scnt;
    if (tid < NWORDS) rem[tid] = 0;
    if (tid == 0) { sflag = 0; sdone = 0; scnt = 0; }
    for (int i = 0; i < NCAND; ++i) {
        __syncthreads();
        if (sdone) break;
        if (tid == 0) {
            float s = ss[(size_t)b * NPAD + i];
            int k = (s > SCORE_TH) && !((rem[i >> 5] >> (i & 31)) & 1u);
            sflag = k;
            if (k) {
                float* row = out + ((size_t)b * DETS + scnt) * 6;
                size_t sbase = ((size_t)b * NPAD + i) * 4;
                row[0] = sb[sbase]; row[1] = sb[sbase + 1];
                row[2] = sb[sbase + 2]; row[3] = sb[sbase + 3];
                row[4] = s; row[5] = (float)sl[(size_t)b * NPAD + i];
                scnt++;
                if (scnt >= DETS) sdone = 1;
            }
        }
        __syncthreads();
        if (sflag && tid < NWORDS)
            rem[tid] |= supp[((size_t)b * NPAD + i) * NWORDS + tid];
    }
    __syncthreads();
    if (tid == 0) {
        for (int r = scnt; r < DETS; ++r) {
            float* row = out + ((size_t)b * DETS + r) * 6;
            row[0] = 0.0f; row[1] = 0.0f; row[2] = 0.0f; row[3] = 0.0f;
            row[4] = -1.0f; row[5] = -1.0f;
        }
    }
}

extern "C" void kernel_launch(void* const* d_in, const int* in_sizes, int n_in,
                              void* d_out, int out_size, void* d_ws, size_t ws_size,
                              hipStream_t stream) {
    (void)in_sizes; (void)n_in; (void)out_size; (void)ws_size;
    const float* an[3] = {(const float*)d_in[0], (const float*)d_in[4], (const float*)d_in[8]};
    const int*   co[3] = {(const int*)d_in[1],   (const int*)d_in[5],   (const int*)d_in[9]};
    const float* lg[3] = {(const float*)d_in[2], (const float*)d_in[6], (const float*)d_in[10]};
    const float* rg[3] = {(const float*)d_in[3], (const float*)d_in[7], (const float*)d_in[11]};
    const float* img   = (const float*)d_in[12];
    static const int Ks[3] = {4096, 2048, 1024};

    char* w = (char*)d_ws;
    size_t o = 0;
    auto alloc = [&](size_t bytes) -> size_t {
        size_t r = o; o += (bytes + 255) & ~(size_t)255; return r;
    };
    size_t o_hist = alloc(24 * NBINS * sizeof(int));
    size_t o_meta = alloc(24 * 4 * sizeof(int));
    size_t o_cnt  = alloc(24 * 2 * sizeof(int));
    size_t o_cand = alloc(24 * 1024 * sizeof(int));
    size_t o_tie  = alloc((size_t)24 * TIE_CAP * sizeof(int));
    size_t o_box  = alloc((size_t)B_IMG * NPAD * 4 * sizeof(float));
    size_t o_sc   = alloc((size_t)B_IMG * NPAD * sizeof(float));
    size_t o_lb   = alloc((size_t)B_IMG * NPAD * sizeof(int));
    size_t o_sb   = alloc((size_t)B_IMG * NPAD * 4 * sizeof(float));
    size_t o_sob  = alloc((size_t)B_IMG * NPAD * 4 * sizeof(float));
    size_t o_ss   = alloc((size_t)B_IMG * NPAD * sizeof(float));
    size_t o_sl   = alloc((size_t)B_IMG * NPAD * sizeof(int));
    size_t o_ar   = alloc((size_t)B_IMG * NPAD * sizeof(float));
    size_t o_supp = alloc((size_t)B_IMG * NPAD * NWORDS * sizeof(unsigned));

    hipMemsetAsync(w + o_hist, 0, 24 * NBINS * sizeof(int), stream);
    hipMemsetAsync(w + o_cnt,  0, 24 * 2 * sizeof(int), stream);
    hipMemsetAsync(w + o_supp, 0, (size_t)B_IMG * NPAD * NWORDS * sizeof(unsigned), stream);

    for (int l = 0; l < 3; ++l) {
        int perImg = Ks[l] * AC;
        hist_kernel<<<dim3(512, 8), 256, 0, stream>>>(
            lg[l], perImg, (int*)(w + o_hist) + (size_t)l * 8 * NBINS);
    }
    thresh_kernel<<<24, 32, 0, stream>>>((int*)(w + o_hist), (int*)(w + o_meta));
    for (int l = 0; l < 3; ++l) {
        int perImg = Ks[l] * AC;
        compact_kernel<<<dim3(512, 8), 256, 0, stream>>>(
            lg[l], perImg, l, (const int*)(w + o_meta), (int*)(w + o_cnt),
            (int*)(w + o_cand), (int*)(w + o_tie));
    }
    tie_kernel<<<24, 1024, 0, stream>>>(
        (const int*)(w + o_meta), (const int*)(w + o_cnt),
        (const int*)(w + o_tie), (int*)(w + o_cand));
    decode_kernel<<<96, 256, 0, stream>>>(
        an[0], an[1], an[2], co[0], co[1], co[2], lg[0], lg[1], lg[2],
        rg[0], rg[1], rg[2], img, (const int*)(w + o_cand),
        (float*)(w + o_box), (float*)(w + o_sc), (int*)(w + o_lb));
    sort_kernel<<<8, 1024, 0, stream>>>(
        (const float*)(w + o_sc), (const float*)(w + o_box), (const int*)(w + o_lb),
        (float*)(w + o_sb), (float*)(w + o_sob), (float*)(w + o_ss),
        (int*)(w + o_sl), (float*)(w + o_ar));
    supp_kernel<<<dim3(NTILES, 8), 1024, 0, stream>>>(
        (const float*)(w + o_sob), (const float*)(w + o_ar), (unsigned*)(w + o_supp));
    resolve_kernel<<<8, 128, 0, stream>>>(
        (const float*)(w + o_ss), (const float*)(w + o_sb), (const int*)(w + o_sl),
        (const unsigned*)(w + o_supp), (float*)d_out);
}